// GATNet_67783173865547
// MI455X (gfx1250) — compile-verified
//
#include <hip/hip_runtime.h>
#include <math.h>

#define N_NODES   20000
#define N_EDGES   320000
#define ET        (N_EDGES + N_NODES)   // with self loops
#define HEADS     4
#define CDIM      64                    // per-head channels
#define HC        256                   // HEADS*CDIM (same for all 3 layers)
#define NEG_SLOPE 0.2f

typedef __attribute__((ext_vector_type(2))) float v2f;
typedef __attribute__((ext_vector_type(8))) float v8f;

// ---------------------------------------------------------------------------
// GEMM: Y[N, 256] = X[N, K] @ W[K, 256], full fp32 via V_WMMA_F32_16X16X4_F32.
// Register-blocked: one wave owns a 16x64 output tile (= one head), so each
// A fragment is reused by 4 WMMAs (4 B fragments / 4 accumulators) per k-step.
// A layout (16x4 f32): lanes 0-15 -> K=0,1 ; lanes 16-31 -> K=2,3 ; M = lane&15
// B layout (4x16 f32): lanes 0-15 -> K=0,1 ; lanes 16-31 -> K=2,3 ; N = lane&15
// C/D layout: VGPR r holds M = r + 8*(lane>=16), N = lane&15
// ---------------------------------------------------------------------------
__global__ void gemm_wmma_f32(const float* __restrict__ X,
                              const float* __restrict__ W,
                              float* __restrict__ Y, int K) {
  const int lane  = threadIdx.x & 31;
  const int wave  = threadIdx.x >> 5;
  const int tm    = blockIdx.x * 4 + wave;           // m-tile (1250 total)
  if (tm * 16 >= N_NODES) return;
  const int tn4   = blockIdx.y;                      // 64-col group (4 total)
  const int mrow  = tm * 16 + (lane & 15);
  const int khalf = (lane >> 4) * 2;                 // 0 or 2

  const float* __restrict__ xrow = X + (size_t)mrow * K + khalf;
  const float* __restrict__ wcol = W + (size_t)khalf * HC + tn4 * 64 + (lane & 15);

  v8f acc0 = {}, acc1 = {}, acc2 = {}, acc3 = {};
#pragma unroll 2
  for (int k = 0; k < K; k += 4) {
    v2f a;
    a.x = xrow[0];
    a.y = xrow[1];
    v2f b0, b1, b2, b3;
    b0.x = wcol[0];   b0.y = wcol[HC];
    b1.x = wcol[16];  b1.y = wcol[HC + 16];
    b2.x = wcol[32];  b2.y = wcol[HC + 32];
    b3.x = wcol[48];  b3.y = wcol[HC + 48];
    acc0 = __builtin_amdgcn_wmma_f32_16x16x4_f32(false, a, false, b0, (short)0, acc0, false, false);
    acc1 = __builtin_amdgcn_wmma_f32_16x16x4_f32(false, a, false, b1, (short)0, acc1, false, false);
    acc2 = __builtin_amdgcn_wmma_f32_16x16x4_f32(false, a, false, b2, (short)0, acc2, false, false);
    acc3 = __builtin_amdgcn_wmma_f32_16x16x4_f32(false, a, false, b3, (short)0, acc3, false, false);
    xrow += 4;
    wcol += (size_t)4 * HC;
  }

  const int mbase = tm * 16 + (lane >> 4) * 8;
  float* __restrict__ yp = Y + tn4 * 64 + (lane & 15);
#pragma unroll
  for (int r = 0; r < 8; ++r) {
    float* row = yp + (size_t)(mbase + r) * HC;
    row[0]  = acc0[r];
    row[16] = acc1[r];
    row[32] = acc2[r];
    row[48] = acc3[r];
  }
}

// ---------------------------------------------------------------------------
// Attention logits: al[n,h] = sum_c h[n,h,c] * att[h,c]
// ---------------------------------------------------------------------------
__global__ void attn_logits(const float* __restrict__ Hl,
                            const float* __restrict__ att_src,
                            const float* __restrict__ att_dst,
                            float* __restrict__ als, float* __restrict__ ald) {
  int idx = blockIdx.x * blockDim.x + threadIdx.x;   // n*HEADS + h
  if (idx >= N_NODES * HEADS) return;
  const int h = idx & 3;
  const float* __restrict__ hp = Hl + (size_t)(idx >> 2) * HC + h * CDIM;
  const float* __restrict__ as = att_src + h * CDIM;
  const float* __restrict__ ad = att_dst + h * CDIM;
  float s0 = 0.f, s1 = 0.f;
#pragma unroll 8
  for (int c = 0; c < CDIM; ++c) {
    float v = hp[c];
    s0 += v * as[c];
    s1 += v * ad[c];
  }
  als[idx] = s0;
  ald[idx] = s1;
}

__global__ void fill_f32(float* __restrict__ p, float v, int n) {
  int i = blockIdx.x * blockDim.x + threadIdx.x;
  if (i < n) p[i] = v;
}

__device__ __forceinline__ int edge_src(const int* __restrict__ ei, int e) {
  return (e < N_EDGES) ? ei[e] : (e - N_EDGES);
}
__device__ __forceinline__ int edge_dst(const int* __restrict__ ei, int e) {
  return (e < N_EDGES) ? ei[N_EDGES + e] : (e - N_EDGES);
}

// Float atomic max via signed-max (>=0) / unsigned-min (<0) bit trick.
__device__ __forceinline__ void atomicMaxF(float* addr, float v) {
  if (v >= 0.f) atomicMax((int*)addr, __float_as_int(v));
  else          atomicMin((unsigned int*)addr, __float_as_uint(v));
}

// Pass 1: alpha = leaky_relu(al_src[src]+al_dst[dst]); segment max into m[dst]
__global__ void edge_alpha_max(const int* __restrict__ ei,
                               const float* __restrict__ als,
                               const float* __restrict__ ald,
                               float* __restrict__ ebuf, float* __restrict__ m) {
  int idx = blockIdx.x * blockDim.x + threadIdx.x;   // e*HEADS + h
  if (idx >= ET * HEADS) return;
  const int e = idx >> 2, h = idx & 3;
  const int s = edge_src(ei, e), d = edge_dst(ei, e);
  float a = als[s * HEADS + h] + ald[d * HEADS + h];
  a = (a > 0.f) ? a : NEG_SLOPE * a;
  ebuf[idx] = a;
  atomicMaxF(&m[d * HEADS + h], a);
}

// Pass 2: e = exp(alpha - m[dst]); segment sum into s[dst]
__global__ void edge_expsum(const int* __restrict__ ei,
                            float* __restrict__ ebuf,
                            const float* __restrict__ m, float* __restrict__ ss) {
  int idx = blockIdx.x * blockDim.x + threadIdx.x;
  if (idx >= ET * HEADS) return;
  const int e = idx >> 2, h = idx & 3;
  const int d = edge_dst(ei, e);
  float v = __expf(ebuf[idx] - m[d * HEADS + h]);
  ebuf[idx] = v;
  atomicAdd(&ss[d * HEADS + h], v);
}

// Pass 3: agg[dst] += (e / s[dst]) * h[src]. 64 threads per edge, 4ch/thread.
__global__ void edge_aggregate(const int* __restrict__ ei,
                               const float* __restrict__ ebuf,
                               const float* __restrict__ ss,
                               const float* __restrict__ Hl,
                               float* __restrict__ agg) {
  int gid = blockIdx.x * blockDim.x + threadIdx.x;
  int e = gid >> 6;
  if (e >= ET) return;
  const int t = gid & 63;
  const int h = t >> 4;
  const int c = (t & 15) * 4;
  const int s = edge_src(ei, e), d = edge_dst(ei, e);
  const float a = ebuf[e * HEADS + h] / (ss[d * HEADS + h] + 1e-16f);
  const float4 hv = *(const float4*)(Hl + (size_t)s * HC + h * CDIM + c);
  float* op = agg + (size_t)d * HC + h * CDIM + c;
  atomicAdd(op + 0, hv.x * a);
  atomicAdd(op + 1, hv.y * a);
  atomicAdd(op + 2, hv.z * a);
  atomicAdd(op + 3, hv.w * a);
}

// y = elu(agg + b), in place (concat layers)
__global__ void bias_elu(float* __restrict__ agg, const float* __restrict__ b, int n) {
  int i = blockIdx.x * blockDim.x + threadIdx.x;
  if (i >= n) return;
  float v = agg[i] + b[i & (HC - 1)];
  agg[i] = (v > 0.f) ? v : expm1f(v);
}

// out[n,c] = mean_h(agg[n,h,c]) + b[c]  (final layer)
__global__ void mean_bias(const float* __restrict__ agg,
                          const float* __restrict__ b, float* __restrict__ out) {
  int i = blockIdx.x * blockDim.x + threadIdx.x;    // n*64 + c
  if (i >= N_NODES * CDIM) return;
  const int n = i >> 6, c = i & 63;
  const float* p = agg + (size_t)n * HC + c;
  out[i] = 0.25f * (p[0] + p[CDIM] + p[2 * CDIM] + p[3 * CDIM]) + b[c];
}

// ---------------------------------------------------------------------------
static void run_gat_layer(const float* xin, int K, const float* W,
                          const float* as, const float* ad, const float* b,
                          const int* ei,
                          float* hl, float* agg, float* eb,
                          float* als, float* ald, float* mm, float* ss,
                          bool concat, float* final_out, const float* final_b,
                          hipStream_t stream) {
  // 1) h = x @ W  (WMMA fp32, 16x64 tile per wave)
  dim3 gg((N_NODES / 16 + 3) / 4, HC / 64);
  gemm_wmma_f32<<<gg, 128, 0, stream>>>(xin, W, hl, K);

  // 2) attention logits
  int nh = N_NODES * HEADS;
  attn_logits<<<(nh + 255) / 256, 256, 0, stream>>>(hl, as, ad, als, ald);

  // 3) init reductions
  fill_f32<<<(nh + 255) / 256, 256, 0, stream>>>(mm, -INFINITY, nh);
  fill_f32<<<(nh + 255) / 256, 256, 0, stream>>>(ss, 0.f, nh);
  fill_f32<<<(N_NODES * HC + 255) / 256, 256, 0, stream>>>(agg, 0.f, N_NODES * HC);

  // 4) segment softmax + aggregation over edges
  int eh = ET * HEADS;
  edge_alpha_max<<<(eh + 255) / 256, 256, 0, stream>>>(ei, als, ald, eb, mm);
  edge_expsum  <<<(eh + 255) / 256, 256, 0, stream>>>(ei, eb, mm, ss);
  edge_aggregate<<<(ET * 64 + 255) / 256, 256, 0, stream>>>(ei, eb, ss, hl, agg);

  // 5) epilogue
  if (concat) {
    bias_elu<<<(N_NODES * HC + 255) / 256, 256, 0, stream>>>(agg, b, N_NODES * HC);
  } else {
    mean_bias<<<(N_NODES * CDIM + 255) / 256, 256, 0, stream>>>(agg, final_b, final_out);
  }
}

extern "C" void kernel_launch(void* const* d_in, const int* in_sizes, int n_in,
                              void* d_out, int out_size, void* d_ws, size_t ws_size,
                              hipStream_t stream) {
  (void)in_sizes; (void)n_in; (void)out_size; (void)ws_size;
  const float* x  = (const float*)d_in[0];
  const int*   ei = (const int*)d_in[1];
  const float* W[3]   = {(const float*)d_in[2], (const float*)d_in[6],  (const float*)d_in[10]};
  const float* AS[3]  = {(const float*)d_in[3], (const float*)d_in[7],  (const float*)d_in[11]};
  const float* AD[3]  = {(const float*)d_in[4], (const float*)d_in[8],  (const float*)d_in[12]};
  const float* B[3]   = {(const float*)d_in[5], (const float*)d_in[9],  (const float*)d_in[13]};

  float* ws  = (float*)d_ws;
  float* bufA = ws;                                  // N*HC
  float* bufB = bufA + (size_t)N_NODES * HC;         // N*HC
  float* hl   = bufB + (size_t)N_NODES * HC;         // N*HC
  float* eb   = hl   + (size_t)N_NODES * HC;         // ET*HEADS
  float* als  = eb   + (size_t)ET * HEADS;           // N*HEADS
  float* ald  = als  + (size_t)N_NODES * HEADS;
  float* mm   = ald  + (size_t)N_NODES * HEADS;
  float* ss   = mm   + (size_t)N_NODES * HEADS;

  // Layer 0: x[N,128] -> bufA[N,256] (elu)
  run_gat_layer(x, 128, W[0], AS[0], AD[0], B[0], ei,
                hl, bufA, eb, als, ald, mm, ss,
                /*concat=*/true, nullptr, nullptr, stream);
  // Layer 1: bufA[N,256] -> bufB[N,256] (elu)
  run_gat_layer(bufA, 256, W[1], AS[1], AD[1], B[1], ei,
                hl, bufB, eb, als, ald, mm, ss,
                /*concat=*/true, nullptr, nullptr, stream);
  // Layer 2: bufB[N,256] -> agg bufA -> mean over heads + bias -> d_out[N,64]
  run_gat_layer(bufB, 256, W[2], AS[2], AD[2], B[2], ei,
                hl, bufA, eb, als, ald, mm, ss,
                /*concat=*/false, (float*)d_out, B[2], stream);
}